// MetaNetLinearizedModel_67705864454400
// MI455X (gfx1250) — compile-verified
//
#include <hip/hip_runtime.h>
#include <hip/hip_bf16.h>

// ---------------------------------------------------------------------------
// MetaNet linearized model for MI455X (gfx1250, wave32, WMMA).
//
// out[b,f] = base[b,f] + sum_t c[b,t] * tangent_t[b,f]
//   z1    = x @ W1^T + b1                  (B x FEAT)   <- WMMA f32 phase 1
//   dz1_t = x @ dW1_t^T + db1_t            (B x FEAT)   <- WMMA f32 phase 1
//   h     = relu(z1); base = h @ W2^T + b2
//   c     = relu(base @ mW1^T + mb1) @ mW2^T + mb2
//   g     = sum_t c[:,t] * (z1>0) * dz1_t              (collapses W2 term)
//   out   = base + g @ W2^T + sum_t c[:,t]*(h @ dW2_t^T + db2_t)
//
// Phase 1 streams 770 MB of weights (HBM-bound: ~33 us at 23.3 TB/s).
// 80 blocks x 32 waves = 2560 waves; unroll-8 gives 16 b64 loads in flight
// per wave (~10.5 MB aggregate MLP) plus one branch-free global_prefetch_b8
// per unrolled body (8 KB ahead, one cacheline per 8 steps).
// ---------------------------------------------------------------------------

#define D_IN   150528
#define FEAT   256
#define HID    64
#define NT     4          // task vectors
#define NB     8          // batch
#define NMAT   5          // W1 + 4 task dW1
#define NWAVE  32         // waves per phase-1 workgroup
#define KC     (D_IN / NWAVE)   // 4704, divisible by 32
#define STEPS  (KC / 4)         // 1176 wmma k-steps per wave (1176 = 8*147)
#define PFD    2048             // prefetch distance in floats (8 KB)

// workspace layout (floats)
#define Z_OFF     0                 // [5][8][256] z1 / dz1_t  (10240)
#define H_OFF     10240             // [8][256] h              (2048)
#define BASE_OFF  12288             // [8][256] base           (2048)
#define COEF_OFF  14336             // [8][4] coefs            (32)
#define G_OFF     14368             // [8][256] g              (2048)

typedef __attribute__((ext_vector_type(2))) float v2f;
typedef __attribute__((ext_vector_type(8))) float v8f;

// ---------------------------------------------------------------------------
// Phase 1: grid = 5*16 blocks, block = 1024 threads (32 waves).
// Block (m, n): matrix m (0=W1, 1..4=dW1_t), N-tile n (16 cols of FEAT).
// Wave w handles K-chunk [w*KC, (w+1)*KC) with v_wmma_f32_16x16x4_f32.
// ---------------------------------------------------------------------------
__global__ void __launch_bounds__(1024)
meta_phase1_wmma(const float* __restrict__ x,
                 const float* __restrict__ W1,
                 const float* __restrict__ b1,
                 const float* __restrict__ dW1,
                 const float* __restrict__ db1,
                 float* __restrict__ ws) {
    __shared__ float red[NWAVE * 256];   // 32 KB: per-wave 16x16 tiles

    const int m    = blockIdx.x >> 4;      // matrix id 0..4
    const int n    = blockIdx.x & 15;      // N-tile 0..15
    const int tid  = threadIdx.x;
    const int w    = tid >> 5;             // wave id 0..31
    const int lane = tid & 31;

    const float* Wm = (m == 0) ? W1 : (dW1 + (size_t)(m - 1) * FEAT * D_IN);

    // WMMA 16x16x4 f32 operand layout (wave32):
    //   A (16x4): lane l holds M=l&15, K = (l>>4)*2 + {0,1} in {VGPR0,VGPR1}
    //   B (4x16): lane l holds N=l&15, K = (l>>4)*2 + {0,1}
    const int r  = lane & 15;              // A row (batch), B row (feature)
    const int kk = (lane >> 4) * 2;        // K sub-offset for this lane half
    const int fRow = n * 16 + r;           // weight row (output feature)

    const bool avalid = (r < NB);
    const float* xrow = x + (size_t)(avalid ? r : 0) * D_IN + kk;
    const float* wrow = Wm + (size_t)fRow * D_IN + kk;
    const size_t k0 = (size_t)w * KC;

    v8f acc = {};
#pragma unroll 8
    for (int s = 0; s < STEPS; ++s) {
        const size_t ko = k0 + 4 * (size_t)s;
        // Period divides the unroll factor -> guard is compile-time constant
        // in each unrolled copy: exactly one straight-line prefetch per body.
        // Speculative prefetch past the allocation end is silently dropped.
        if ((s & 7) == 0)
            __builtin_prefetch(wrow + ko + PFD, 0, 1);   // global_prefetch_b8
        v2f av = *(const v2f*)(xrow + ko);   // global_load_b64 (L2-resident x)
        v2f bv = *(const v2f*)(wrow + ko);   // global_load_b64 (HBM stream)
        v2f a;
        a.x = avalid ? av.x : 0.0f;          // zero-pad batch rows 8..15
        a.y = avalid ? av.y : 0.0f;
        acc = __builtin_amdgcn_wmma_f32_16x16x4_f32(
                  false, a, false, bv, (short)0, acc, false, false);
    }

    // Scatter the 16x16 tile to LDS: element (M,N) -> lane ((M>>3)*16 + N), VGPR M&7
#pragma unroll
    for (int j = 0; j < 8; ++j) {
        const int row = j + (lane >> 4) * 8;   // M
        red[w * 256 + row * 16 + (lane & 15)] = acc[j];
    }
    __syncthreads();

    // Deterministic fixed-order reduction over the 32 waves; only rows 0..7 matter.
    if (tid < NB * 16) {
        const int b   = tid >> 4;
        const int col = tid & 15;
        float s = 0.0f;
#pragma unroll
        for (int ww = 0; ww < NWAVE; ++ww)
            s += red[ww * 256 + b * 16 + col];
        const int f = n * 16 + col;
        const float bias = (m == 0) ? b1[f] : db1[(m - 1) * FEAT + f];
        ws[Z_OFF + m * (NB * FEAT) + b * FEAT + f] = s + bias;
    }
}

// ---------------------------------------------------------------------------
// Phase 2a: single block, 256 threads. h, base, MetaNet coefs, g.
// ---------------------------------------------------------------------------
__global__ void __launch_bounds__(256)
meta_phase2a(const float* __restrict__ W2,  const float* __restrict__ b2,
             const float* __restrict__ mW1, const float* __restrict__ mb1,
             const float* __restrict__ mW2, const float* __restrict__ mb2,
             float* __restrict__ ws) {
    __shared__ float sh_h[NB * FEAT];      // 8 KB
    __shared__ float sh_base[NB * FEAT];   // 8 KB
    __shared__ float sh_hid[NB * HID];     // 2 KB
    __shared__ float sh_coef[NB * NT];

    const int tid = threadIdx.x;
    const float* z1 = ws + Z_OFF;          // matrix 0 slot

    for (int i = tid; i < NB * FEAT; i += 256) {
        const float h = fmaxf(z1[i], 0.0f);
        sh_h[i] = h;
        ws[H_OFF + i] = h;
    }
    __syncthreads();

    for (int i = tid; i < NB * FEAT; i += 256) {
        const int b = i >> 8, f = i & 255;
        float s = b2[f];
        const float* hb = sh_h + b * FEAT;
        const float* wr = W2 + (size_t)f * FEAT;
        for (int k = 0; k < FEAT; ++k) s += hb[k] * wr[k];
        sh_base[i] = s;
        ws[BASE_OFF + i] = s;
    }
    __syncthreads();

    for (int i = tid; i < NB * HID; i += 256) {
        const int b = i >> 6, j = i & 63;
        float s = mb1[j];
        for (int f = 0; f < FEAT; ++f) s += sh_base[b * FEAT + f] * mW1[j * FEAT + f];
        sh_hid[i] = fmaxf(s, 0.0f);
    }
    __syncthreads();

    if (tid < NB * NT) {
        const int b = tid >> 2, t = tid & 3;
        float s = mb2[t];
        for (int j = 0; j < HID; ++j) s += sh_hid[b * HID + j] * mW2[t * HID + j];
        sh_coef[tid] = s;
        ws[COEF_OFF + tid] = s;
    }
    __syncthreads();

    // g[b,k] = (z1[b,k] > 0) * sum_t c[b,t] * dz1_t[b,k]
    for (int i = tid; i < NB * FEAT; i += 256) {
        const int b = i >> 8;
        float g = 0.0f;
        if (z1[i] > 0.0f) {
#pragma unroll
            for (int t = 0; t < NT; ++t)
                g += sh_coef[b * NT + t] * ws[Z_OFF + (1 + t) * (NB * FEAT) + i];
        }
        ws[G_OFF + i] = g;
    }
}

// ---------------------------------------------------------------------------
// Phase 2b: grid = 256 (one per output feature f), block = 256 = 8 waves.
// Wave b computes out[b,f]; lanes split K=256, wave32 shfl_xor reduce.
// ---------------------------------------------------------------------------
__global__ void __launch_bounds__(256)
meta_phase2b(const float* __restrict__ ws,
             const float* __restrict__ W2,
             const float* __restrict__ dW2,
             const float* __restrict__ db2,
             float* __restrict__ out) {
    const int f    = blockIdx.x;
    const int tid  = threadIdx.x;
    const int b    = tid >> 5;
    const int lane = tid & 31;

    const float* H  = ws + H_OFF + b * FEAT;
    const float* G  = ws + G_OFF + b * FEAT;
    const float* wr = W2 + (size_t)f * FEAT;
    const float* d0 = dW2 + 0 * FEAT * FEAT + (size_t)f * FEAT;
    const float* d1 = dW2 + 1 * FEAT * FEAT + (size_t)f * FEAT;
    const float* d2 = dW2 + 2 * FEAT * FEAT + (size_t)f * FEAT;
    const float* d3 = dW2 + 3 * FEAT * FEAT + (size_t)f * FEAT;

    float pw = 0.0f, s0 = 0.0f, s1 = 0.0f, s2 = 0.0f, s3 = 0.0f;
    for (int k = lane; k < FEAT; k += 32) {
        const float hk = H[k];
        pw += G[k] * wr[k];
        s0 += hk * d0[k];
        s1 += hk * d1[k];
        s2 += hk * d2[k];
        s3 += hk * d3[k];
    }
#pragma unroll
    for (int off = 16; off > 0; off >>= 1) {
        pw += __shfl_xor(pw, off, 32);
        s0 += __shfl_xor(s0, off, 32);
        s1 += __shfl_xor(s1, off, 32);
        s2 += __shfl_xor(s2, off, 32);
        s3 += __shfl_xor(s3, off, 32);
    }
    if (lane == 0) {
        const float* cf = ws + COEF_OFF + b * NT;
        float r = ws[BASE_OFF + b * FEAT + f] + pw;
        r += cf[0] * (s0 + db2[0 * FEAT + f]);
        r += cf[1] * (s1 + db2[1 * FEAT + f]);
        r += cf[2] * (s2 + db2[2 * FEAT + f]);
        r += cf[3] * (s3 + db2[3 * FEAT + f]);
        out[b * FEAT + f] = r;
    }
}

// ---------------------------------------------------------------------------
extern "C" void kernel_launch(void* const* d_in, const int* in_sizes, int n_in,
                              void* d_out, int out_size, void* d_ws, size_t ws_size,
                              hipStream_t stream) {
    const float* x   = (const float*)d_in[0];
    const float* W1  = (const float*)d_in[1];
    const float* b1  = (const float*)d_in[2];
    const float* W2  = (const float*)d_in[3];
    const float* b2  = (const float*)d_in[4];
    const float* mW1 = (const float*)d_in[5];
    const float* mb1 = (const float*)d_in[6];
    const float* mW2 = (const float*)d_in[7];
    const float* mb2 = (const float*)d_in[8];
    const float* dW1 = (const float*)d_in[9];
    const float* db1 = (const float*)d_in[10];
    const float* dW2 = (const float*)d_in[11];
    const float* db2 = (const float*)d_in[12];
    float* ws  = (float*)d_ws;
    float* out = (float*)d_out;

    meta_phase1_wmma<<<NMAT * 16, NWAVE * 32, 0, stream>>>(x, W1, b1, dW1, db1, ws);
    meta_phase2a<<<1, 256, 0, stream>>>(W2, b2, mW1, mb1, mW2, mb2, ws);
    meta_phase2b<<<FEAT, 256, 0, stream>>>(ws, W2, dW2, db2, out);
}